// rpnProposalLayer_47407849013557
// MI455X (gfx1250) — compile-verified
//
#include <hip/hip_runtime.h>

// ---------------- problem constants ----------------
#define IMG      1024
#define NB       4          // batch
#define NL       5          // levels
#define TOTAL_A  261888     // sum of per-level anchors (divisible by 16)
#define PRE_NMS  2000
#define POST_NMS 1000
#define KPAD     2048       // padded per-level candidate count (pow2 for sort)
#define CCAP     5000       // per-image concat region (4768 used)
#define NMS_TH   0.7f

typedef __attribute__((ext_vector_type(2))) float v2f;
typedef __attribute__((ext_vector_type(8))) float v8f;

__device__ __constant__ int c_A[NL]   = {196608, 49152, 12288, 3072, 768};
__device__ __constant__ int c_off[NL] = {0, 196608, 245760, 258048, 261120};

static __device__ __forceinline__ int imin(int a, int b) { return a < b ? a : b; }

// CDNA5 async global->LDS copy of 16 bytes (tracked by ASYNCcnt)
static __device__ __forceinline__ void async_copy_b128(void* lds_dst, const void* gsrc) {
    unsigned lo = (unsigned)(unsigned long long)lds_dst;   // low 32 bits = group-segment offset
    asm volatile("global_load_async_to_lds_b128 %0, %1, off"
                 :: "v"(lo), "v"(gsrc) : "memory");
}
static __device__ __forceinline__ void async_wait0() {
    asm volatile("s_wait_asynccnt 0" ::: "memory");
}

// ---------------- kernel 1: anchor decode; corner transform via V_WMMA_F32_16X16X4_F32 ----------------
// Each wave32 decodes a tile of 16 anchors: center-form [ncx,ncy,nw,nh] (16x4 A-matrix)
// times constant corner map Mt embedded in a 4x16 B-matrix -> corners in 16x16 D.
__global__ __launch_bounds__(256) void decode_kernel(
    const float* __restrict__ p0, const float* __restrict__ r0,
    const float* __restrict__ p1, const float* __restrict__ r1,
    const float* __restrict__ p2, const float* __restrict__ r2,
    const float* __restrict__ p3, const float* __restrict__ r3,
    const float* __restrict__ p4, const float* __restrict__ r4,
    const float* __restrict__ info,
    float* __restrict__ scores, float* __restrict__ boxes)
{
    const int lane = threadIdx.x & 31;
    const int wv   = threadIdx.x >> 5;
    const int t0   = (blockIdx.x * 8 + wv) * 16;   // first anchor of this wave's tile (exact grid)
    const int m    = lane & 15;                    // box within tile this lane computes
    const int t    = t0 + m;
    const int b    = t / TOTAL_A;
    const int a    = t - b * TOTAL_A;

    int l, off;
    if      (a < 196608) { l = 0; off = 0; }
    else if (a < 245760) { l = 1; off = 196608; }
    else if (a < 258048) { l = 2; off = 245760; }
    else if (a < 261120) { l = 3; off = 258048; }
    else                 { l = 4; off = 261120; }

    const float* P[NL] = {p0, p1, p2, p3, p4};
    const float* R[NL] = {r0, r1, r2, r3, r4};

    int al   = a - off;
    int g    = 256 >> l;                 // grid size per side
    int cell = al / 3;
    int rz   = al - cell * 3;
    int iy   = cell / g;
    int ix   = cell - iy * g;

    float stride = (float)(4 << l);
    float cx = (ix + 0.5f) * stride;
    float cy = (iy + 0.5f) * stride;
    float s8 = (float)(32 << l) * 8.0f;  // size_base * scale
    // ratios 0.5, 1.0, 2.0 -> w = s8*sqrt(1/r), h = s8*sqrt(r)
    float w = (rz == 0) ? s8 * 1.41421356f : ((rz == 1) ? s8 : s8 * 0.70710678f);
    float h = (rz == 0) ? s8 * 0.70710678f : ((rz == 1) ? s8 : s8 * 1.41421356f);

    int Al = g * g * 3;
    const float* pr = P[l] + ((size_t)b * Al + al) * 2;
    const float* rg = R[l] + ((size_t)b * Al + al) * 4;

    float s  = pr[0];
    float dx = rg[0], dy = rg[1], dw = rg[2], dh = rg[3];

    float ncx = dx * w + cx;
    float ncy = dy * h + cy;
    float nw  = __expf(dw) * w;
    float nh  = __expf(dh) * h;

    // A-matrix (16x4 f32, 2 VGPRs): lanes 0-15 carry K=0,1 = (ncx,ncy) of box M=lane;
    // lanes 16-31 carry K=2,3 = (nw,nh) of box M=lane-16.
    v2f A;
    if (lane < 16) { A.x = ncx; A.y = ncy; }
    else           { A.x = nw;  A.y = nh;  }

    // B-matrix (4x16 f32, 2 VGPRs): corner map Mt in columns 0-3, zero elsewhere.
    //   K0 (ncx): [ 1, 0, 1, 0]   K1 (ncy): [0,  1, 0,  1]
    //   K2 (nw) : [-.5,0,.5, 0]   K3 (nh) : [0,-.5, 0,.5]
    const int n = lane & 15;
    v2f Bv; Bv.x = 0.0f; Bv.y = 0.0f;
    if (n < 4) {
        if (lane < 16) {  // rows K=0 (x-center), K=1 (y-center)
            Bv.x = (n == 0 || n == 2) ? 1.0f : 0.0f;
            Bv.y = (n == 1 || n == 3) ? 1.0f : 0.0f;
        } else {          // rows K=2 (width), K=3 (height)
            Bv.x = (n == 0) ? -0.5f : ((n == 2) ? 0.5f : 0.0f);
            Bv.y = (n == 1) ? -0.5f : ((n == 3) ? 0.5f : 0.0f);
        }
    }

    v8f D = {};
    D = __builtin_amdgcn_wmma_f32_16x16x4_f32(false, A, false, Bv,
                                              (short)0, D, false, false);

    if (lane < 16) scores[(size_t)t] = s;

    // D layout: lane L, VGPR r = corner component n=L%16 of box m2 = r + (L<16 ? 0 : 8).
    float Hc = info[b * 3 + 0] - 1.0f;
    float Wc = info[b * 3 + 1] - 1.0f;
    float lim = (n == 1 || n == 3) ? Hc : Wc;
    if (n < 4) {
        int mbase = (lane < 16) ? 0 : 8;
#pragma unroll
        for (int r = 0; r < 8; ++r) {
            float v = D[r];
            v = fminf(fmaxf(v, 0.0f), lim);
            boxes[(size_t)(t0 + mbase + r) * 4 + n] = v;
        }
    }
}

// ---------------- kernel 2: per-(b,l) histogram-select + bitonic top-2048 ----------------
__global__ __launch_bounds__(1024) void topk_kernel(
    const float* __restrict__ scores, const float* __restrict__ boxes,
    float* __restrict__ tsc, float* __restrict__ tbx)
{
    const int blk = blockIdx.x;          // b*5 + l
    const int b = blk / NL, l = blk - b * NL;
    const int Al  = c_A[l];
    const int off = c_off[l];
    const int NK  = imin(PRE_NMS, Al);
    const int tid = threadIdx.x;

    __shared__ unsigned hist[4096];
    __shared__ float    cs[4096];
    __shared__ int      ci[4096];
    __shared__ unsigned cnt;
    __shared__ int      thrBin;

    const float* ls = scores + (size_t)b * TOTAL_A + off;

    for (int i = tid; i < 4096; i += 1024) hist[i] = 0u;
    if (tid == 0) cnt = 0u;
    __syncthreads();

    for (int i = tid; i < Al; i += 1024) {
        float s = ls[i];                         // scores are in [0,1)
        int bin = (int)(s * 4096.0f);
        bin = bin < 0 ? 0 : (bin > 4095 ? 4095 : bin);
        atomicAdd(&hist[bin], 1u);
    }
    __syncthreads();

    if (tid == 0) {
        unsigned target = (unsigned)imin(KPAD, Al);
        unsigned acc = 0; int t = 0;
        for (int i = 4095; i >= 0; --i) {
            acc += hist[i];
            if (acc >= target) { t = i; break; }
        }
        thrBin = t;
    }
    __syncthreads();

    float thr = (float)thrBin * (1.0f / 4096.0f);
    for (int i = tid; i < Al; i += 1024) {
        float s = ls[i];
        if (s >= thr) {
            unsigned p = atomicAdd(&cnt, 1u);
            if (p < 4096u) { cs[p] = s; ci[p] = i; }
        }
    }
    __syncthreads();

    unsigned nvalid = cnt; if (nvalid > 4096u) nvalid = 4096u;
    for (int i = tid; i < 4096; i += 1024)
        if ((unsigned)i >= nvalid) { cs[i] = -1e30f; ci[i] = -1; }
    __syncthreads();

    // bitonic sort, descending
    for (int k = 2; k <= 4096; k <<= 1) {
        for (int j = k >> 1; j > 0; j >>= 1) {
            for (int i = tid; i < 4096; i += 1024) {
                int ixj = i ^ j;
                if (ixj > i) {
                    float a = cs[i], bb = cs[ixj];
                    bool sw = ((i & k) == 0) ? (a < bb) : (a > bb);
                    if (sw) {
                        cs[i] = bb; cs[ixj] = a;
                        int tt = ci[i]; ci[i] = ci[ixj]; ci[ixj] = tt;
                    }
                }
            }
            __syncthreads();
        }
    }

    const float4* lb  = (const float4*)(boxes + ((size_t)b * TOTAL_A + off) * 4);
    float*  osc = tsc + (size_t)blk * KPAD;
    float4* obx = (float4*)(tbx + (size_t)blk * KPAD * 4);
    for (int i = tid; i < KPAD; i += 1024) {
        float s = cs[i]; int idx = ci[i];
        bool valid = (i < NK) && (idx >= 0) && (s > -1e29f);
        osc[i] = valid ? s : -1e30f;
        obx[i] = valid ? lb[idx] : make_float4(0, 0, 0, 0);
    }
}

// ---------------- kernel 3: greedy NMS on LDS-resident tile (async staged) ----------------
__global__ __launch_bounds__(1024) void nms_kernel(
    const float* __restrict__ tsc, const float* __restrict__ tbx,
    int* __restrict__ keep)
{
    const int blk = blockIdx.x;
    const int tid = threadIdx.x;

    __shared__ float sbx[KPAD * 4];   // 32 KB of boxes, async-loaded
    __shared__ float sar[KPAD];
    __shared__ int   sk[KPAD];

    const float* gb = tbx + (size_t)blk * KPAD * 4;
    for (int r = tid; r < KPAD; r += 1024)
        async_copy_b128(&sbx[r * 4], gb + (size_t)r * 4);   // ASYNCcnt path
    async_wait0();
    __syncthreads();

    const float* gs = tsc + (size_t)blk * KPAD;
    for (int r = tid; r < KPAD; r += 1024) {
        float x1 = sbx[r * 4], y1 = sbx[r * 4 + 1];
        float x2 = sbx[r * 4 + 2], y2 = sbx[r * 4 + 3];
        sar[r] = fmaxf(x2 - x1, 0.0f) * fmaxf(y2 - y1, 0.0f);
        sk[r]  = (gs[r] > -1e29f) ? 1 : 0;
    }
    __syncthreads();

    for (int i = 0; i < KPAD - 1; ++i) {
        if (sk[i]) {   // uniform branch: all threads read same LDS word
            float x1i = sbx[i * 4],     y1i = sbx[i * 4 + 1];
            float x2i = sbx[i * 4 + 2], y2i = sbx[i * 4 + 3];
            float ai  = sar[i];
            for (int j = i + 1 + tid; j < KPAD; j += 1024) {
                if (sk[j]) {
                    float xx1 = fmaxf(x1i, sbx[j * 4]);
                    float yy1 = fmaxf(y1i, sbx[j * 4 + 1]);
                    float xx2 = fminf(x2i, sbx[j * 4 + 2]);
                    float yy2 = fminf(y2i, sbx[j * 4 + 3]);
                    float iw = fmaxf(xx2 - xx1, 0.0f);
                    float ih = fmaxf(yy2 - yy1, 0.0f);
                    float inter = iw * ih;
                    float iou = inter / fmaxf(ai + sar[j] - inter, 1e-9f);
                    if (iou > NMS_TH) sk[j] = 0;
                }
            }
        }
        __syncthreads();
    }

    int* gk = keep + (size_t)blk * KPAD;
    for (int r = tid; r < KPAD; r += 1024) gk[r] = sk[r];
}

// ---------------- kernel 4: stable compaction of survivors into concat arrays ----------------
__global__ __launch_bounds__(1024) void compact_kernel(
    const float* __restrict__ tsc, const float* __restrict__ tbx,
    const int* __restrict__ keep,
    float* __restrict__ csc, float* __restrict__ cbx)
{
    const int blk = blockIdx.x;
    const int b = blk / NL, l = blk - b * NL;
    const int tid = threadIdx.x;
    const int NK   = imin(PRE_NMS, c_A[l]);
    const int CAP  = imin(POST_NMS, NK);       // 1000,1000,1000,1000,768
    const int cOff = 1000 * l;

    __shared__ int sk[KPAD];
    __shared__ int pos[KPAD];

    const int* gk = keep + (size_t)blk * KPAD;
    for (int r = tid; r < KPAD; r += 1024) sk[r] = gk[r];
    __syncthreads();

    if (tid == 0) {
        int c = 0;
        for (int r = 0; r < KPAD; ++r) { pos[r] = c; c += sk[r]; }
    }
    // init output segment: score = -1 (matches reference masking), box = 0
    for (int o = tid; o < CAP; o += 1024) {
        csc[(size_t)b * CCAP + cOff + o] = -1.0f;
        ((float4*)cbx)[(size_t)b * CCAP + cOff + o] = make_float4(0, 0, 0, 0);
    }
    __syncthreads();

    const float*  gsc = tsc + (size_t)blk * KPAD;
    const float4* gbx = (const float4*)(tbx + (size_t)blk * KPAD * 4);
    for (int r = tid; r < NK; r += 1024) {
        if (sk[r] && pos[r] < CAP) {
            int o = cOff + pos[r];
            csc[(size_t)b * CCAP + o] = gsc[r];
            ((float4*)cbx)[(size_t)b * CCAP + o] = gbx[r];
        }
    }
}

// ---------------- kernel 5: per-image final top-1000 + emit ----------------
__global__ __launch_bounds__(1024) void final_kernel(
    const float* __restrict__ csc, const float* __restrict__ cbx,
    float* __restrict__ out)
{
    const int b = blockIdx.x;
    const int tid = threadIdx.x;

    __shared__ float          fsv[8192];
    __shared__ unsigned short fid[8192];

    const float* gs = csc + (size_t)b * CCAP;
    for (int r = tid; r < 1192; r += 1024)            // 1192*4 = 4768 scores
        async_copy_b128(&fsv[r * 4], gs + r * 4);
    for (int i = tid; i < 8192; i += 1024)
        fid[i] = (unsigned short)(i < 4768 ? i : 0);
    async_wait0();
    __syncthreads();
    for (int i = 4768 + tid; i < 8192; i += 1024) fsv[i] = -1e30f;
    __syncthreads();

    for (int k = 2; k <= 8192; k <<= 1) {
        for (int j = k >> 1; j > 0; j >>= 1) {
            for (int i = tid; i < 8192; i += 1024) {
                int ixj = i ^ j;
                if (ixj > i) {
                    float a = fsv[i], bb = fsv[ixj];
                    bool sw = ((i & k) == 0) ? (a < bb) : (a > bb);
                    if (sw) {
                        fsv[i] = bb; fsv[ixj] = a;
                        unsigned short t = fid[i]; fid[i] = fid[ixj]; fid[ixj] = t;
                    }
                }
            }
            __syncthreads();
        }
    }

    const float4* gb = (const float4*)(cbx + (size_t)b * CCAP * 4);
    for (int k2 = tid; k2 < POST_NMS; k2 += 1024) {
        float s = fsv[k2];
        int idx = fid[k2];
        float4 bv = (s >= 0.0f) ? gb[idx] : make_float4(0, 0, 0, 0);
        float* o = out + (size_t)(b * POST_NMS + k2) * 5;
        o[0] = (float)b; o[1] = bv.x; o[2] = bv.y; o[3] = bv.z; o[4] = bv.w;
    }
}

// ---------------- host launcher ----------------
extern "C" void kernel_launch(void* const* d_in, const int* in_sizes, int n_in,
                              void* d_out, int out_size, void* d_ws, size_t ws_size,
                              hipStream_t stream)
{
    (void)in_sizes; (void)n_in; (void)out_size; (void)ws_size;
    // setup_inputs() dict order: probs0, reg0, probs1, reg1, ..., probs4, reg4, img_info
    const float* p0 = (const float*)d_in[0];
    const float* r0 = (const float*)d_in[1];
    const float* p1 = (const float*)d_in[2];
    const float* r1 = (const float*)d_in[3];
    const float* p2 = (const float*)d_in[4];
    const float* r2 = (const float*)d_in[5];
    const float* p3 = (const float*)d_in[6];
    const float* r3 = (const float*)d_in[7];
    const float* p4 = (const float*)d_in[8];
    const float* r4 = (const float*)d_in[9];
    const float* info = (const float*)d_in[10];
    float* out = (float*)d_out;

    // workspace carve-up (256B aligned)
    char* base = (char*)d_ws;
    size_t o = 0;
    auto carve = [&](size_t bytes) -> void* {
        void* r = base + o;
        o = (o + bytes + 255) & ~(size_t)255;
        return r;
    };
    float* scores = (float*)carve(sizeof(float) * NB * TOTAL_A);
    float* boxes  = (float*)carve(sizeof(float) * NB * TOTAL_A * 4);
    float* tsc    = (float*)carve(sizeof(float) * NB * NL * KPAD);
    float* tbx    = (float*)carve(sizeof(float) * NB * NL * KPAD * 4);
    int*   keep   = (int*)  carve(sizeof(int)   * NB * NL * KPAD);
    float* csc    = (float*)carve(sizeof(float) * NB * CCAP);
    float* cbx    = (float*)carve(sizeof(float) * NB * CCAP * 4);

    // decode: one wave per 16 anchors -> 128 anchors per 256-thread block; exact grid
    const int nblocks = (NB * TOTAL_A) / 128;   // 1047552 / 128 = 8184, no remainder
    decode_kernel<<<nblocks, 256, 0, stream>>>(
        p0, r0, p1, r1, p2, r2, p3, r3, p4, r4, info, scores, boxes);
    topk_kernel<<<NB * NL, 1024, 0, stream>>>(scores, boxes, tsc, tbx);
    nms_kernel<<<NB * NL, 1024, 0, stream>>>(tsc, tbx, keep);
    compact_kernel<<<NB * NL, 1024, 0, stream>>>(tsc, tbx, keep, csc, cbx);
    final_kernel<<<NB, 1024, 0, stream>>>(csc, cbx, out);
}